// EncodeLayer_81741817577736
// MI455X (gfx1250) — compile-verified
//
#include <hip/hip_runtime.h>

#define TPB 256

static constexpr int GT_M  = 100;        // GT rows per batch (reference M)
static constexpr int MROWS = GT_M + 1;   // +1 prepended zero row

#if __has_builtin(__builtin_amdgcn_global_load_async_to_lds_b128)
#define USE_ASYNC_LDS 1
#else
#define USE_ASYNC_LDS 0
#endif

typedef int v4i __attribute__((vector_size(16)));
typedef __attribute__((address_space(1))) v4i* global_v4i_p;
typedef __attribute__((address_space(3))) v4i* shared_v4i_p;

__global__ __launch_bounds__(TPB) void encode_ssd_kernel(
    const float* __restrict__ labels,    // [B, GT_M, 5]
    const float* __restrict__ dboxes,    // [N, 4]
    float* __restrict__ out_labeled,     // [B, N, 5]
    float* __restrict__ out_ious,        // [B, MROWS, N]
    int N)
{
    const int tid = threadIdx.x;
    const int b   = blockIdx.y;
    const int n0  = blockIdx.x * TPB;
    const int n   = n0 + tid;
    const bool active = (n < N);

    __shared__ __align__(16) float s_raw[GT_M * 5];          // 2000 B
    __shared__ float s_x1[MROWS], s_y1[MROWS], s_x2[MROWS], s_y2[MROWS];
    __shared__ float s_area[MROWS], s_cls[MROWS], s_val[MROWS];
    __shared__ float s_stage[TPB * 5];

    // ---- Stage this batch's GT rows into LDS (CDNA5 async-to-LDS path) ----
    const float* gsrc = labels + (size_t)b * (GT_M * 5);
#if USE_ASYNC_LDS
    if (tid < (GT_M * 5) / 4) {   // 125 lanes x B128 = 2000 bytes
        __builtin_amdgcn_global_load_async_to_lds_b128(
            (global_v4i_p)(gsrc + tid * 4),
            (shared_v4i_p)(&s_raw[tid * 4]),
            /*imm offset*/0, /*cpol*/0);
    }
#if __has_builtin(__builtin_amdgcn_s_wait_asynccnt)
    __builtin_amdgcn_s_wait_asynccnt(0);
#else
    asm volatile("s_wait_asynccnt 0" ::: "memory");
#endif
#else
    for (int i = tid; i < GT_M * 5; i += TPB) s_raw[i] = gsrc[i];
#endif
    __syncthreads();

    // ---- Build SoA for the 101 rows (row 0 is the zero pad row) ----
    if (tid < MROWS) {
        float x1 = 0.f, y1 = 0.f, x2 = 0.f, y2 = 0.f, c = 0.f;
        if (tid > 0) {
            const float* r = &s_raw[(tid - 1) * 5];
            x1 = r[0]; y1 = r[1]; x2 = r[2]; y2 = r[3]; c = r[4];
        }
        s_x1[tid] = x1; s_y1[tid] = y1; s_x2[tid] = x2; s_y2[tid] = y2;
        s_area[tid] = (x2 - x1) * (y2 - y1);
        s_cls[tid]  = c;
        s_val[tid]  = (c != 0.0f) ? 1.0f : 0.0f;
    }
    __syncthreads();

    // ---- Per-anchor IoU sweep + running argmax (first-max-wins) ----
    float ax1 = 0.f, ay1 = 0.f, ax2 = 1.f, ay2 = 1.f;   // dummy box for inactive lanes
    if (active) {
        const float4 db = reinterpret_cast<const float4*>(dboxes)[n];
        ax1 = db.x; ay1 = db.y; ax2 = db.z; ay2 = db.w;
    }
    const float area_a = (ax2 - ax1) * (ay2 - ay1);

    float best = -1.0f;
    int   bidx = 0;
    float* ious_bn = out_ious + (size_t)b * MROWS * N + n;
#pragma unroll 4
    for (int m = 0; m < MROWS; ++m) {
        float ix1 = fmaxf(s_x1[m], ax1);
        float iy1 = fmaxf(s_y1[m], ay1);
        float ix2 = fminf(s_x2[m], ax2);
        float iy2 = fminf(s_y2[m], ay2);
        float iw  = fmaxf(ix2 - ix1, 0.0f);
        float ih  = fmaxf(iy2 - iy1, 0.0f);
        float inter = iw * ih;
        float iou = (inter / (s_area[m] + area_a - inter)) * s_val[m];
        if (active) {
            __builtin_nontemporal_store(iou, ious_bn + (size_t)m * N);  // streaming, never re-read
            if (iou > best) { best = iou; bidx = m; }
        }
    }

    // ---- Gather winner, compute regression offsets ----
    const float pos = (best > 0.5f) ? 1.0f : 0.0f;
    const int   idx = (pos > 0.0f) ? bidx : 0;
    const float gcls = s_cls[idx];

    float bx1, by1, bx2, by2;
    if (gcls == 0.0f) { bx1 = ax1; by1 = ay1; bx2 = ax2; by2 = ay2; }  // unmatched -> anchor itself
    else              { bx1 = s_x1[idx]; by1 = s_y1[idx]; bx2 = s_x2[idx]; by2 = s_y2[idx]; }

    const float a_cx = (ax1 + ax2) * 0.5f, a_cy = (ay1 + ay2) * 0.5f;
    const float a_w  = ax2 - ax1,          a_h  = ay2 - ay1;
    const float t_cx = (bx1 + bx2) * 0.5f, t_cy = (by1 + by2) * 0.5f;
    const float t_w  = bx2 - bx1,          t_h  = by2 - by1;

    const float ox = (t_cx - a_cx) / a_w;
    const float oy = (t_cy - a_cy) / a_h;
    const float ow = __logf(t_w / a_w);
    const float oh = __logf(t_h / a_h);

    // ---- Coalesce [n][5] writes through LDS ----
    __syncthreads();
    if (active) {
        float* s = &s_stage[tid * 5];     // stride 5 (odd) -> conflict-free across a wave32
        s[0] = ox; s[1] = oy; s[2] = ow; s[3] = oh; s[4] = gcls * pos;
    }
    __syncthreads();
    const int cnt = (N - n0 < TPB) ? (N - n0) : TPB;
    float* dst = out_labeled + ((size_t)b * N + n0) * 5;
    for (int i = tid; i < cnt * 5; i += TPB)
        __builtin_nontemporal_store(s_stage[i], dst + i);
}

extern "C" void kernel_launch(void* const* d_in, const int* in_sizes, int n_in,
                              void* d_out, int out_size, void* d_ws, size_t ws_size,
                              hipStream_t stream) {
    const float* labels = (const float*)d_in[0];   // [B, 100, 5]
    const float* dboxes = (const float*)d_in[1];   // [N, 4]
    const int N = in_sizes[1] / 4;
    const int B = in_sizes[0] / (GT_M * 5);

    float* out_labeled = (float*)d_out;                       // [B, N, 5]
    float* out_ious    = out_labeled + (size_t)B * N * 5;     // [B, 101, N]

    dim3 grid((N + TPB - 1) / TPB, B);
    encode_ssd_kernel<<<grid, TPB, 0, stream>>>(labels, dboxes, out_labeled, out_ious, N);
}